// FFTEncDec_26439818674333
// MI455X (gfx1250) — compile-verified
//
#include <hip/hip_runtime.h>
#include <cstddef>

// ---------------- compile-time mask (matches numpy linspace semantics) ------
constexpr int Hc = 64, Wc = 64, Wr = Wc / 2 + 1;      // 64 x 33 rfft grid
constexpr int Sd = 16, Pd = 256, Cd = 3, ENC = 1024;
constexpr int MAXK = Hc * Wr;

struct MaskIdx {
    int K;
    short iy[MAXK];
    short ix[MAXK];
};

constexpr MaskIdx make_mask() {
    MaskIdx m{};
    int k = 0;
    const double ystep = 2.0 / 63.0;   // np.linspace(-1,1,64) step
    const double xstep = 1.0 / 32.0;   // np.linspace(0,1,33) step
    for (int i = 0; i < Hc; ++i) {
        double y = -1.0 + i * ystep;
        for (int j = 0; j < Wr; ++j) {
            double x = j * xstep;
            bool in = (x < 0.1) || (y < 0.1 && y > -0.1) || (x * x + y * y <= 0.4 * 0.4);
            if (in) { m.iy[k] = (short)i; m.ix[k] = (short)j; ++k; }
        }
    }
    m.K = k;
    return m;
}

constexpr MaskIdx H_MASK = make_mask();
constexpr int K      = H_MASK.K;          // mask points per (s,p,c)
constexpr int MD     = 6 * K;             // feature dim = real+imag, 3 channels
constexpr int KPAD16 = (MD + 15) & ~15;   // feats K-dim padded to K-step of 16
constexpr int KFULL  = MD & ~15;          // full (unguarded) K-steps

__constant__ MaskIdx c_mask = make_mask();

typedef __attribute__((ext_vector_type(2))) float v2f;
typedef __attribute__((ext_vector_type(8))) float v8f;

__device__ __forceinline__ v8f wmma4(v2f a, v2f b, v8f c) {
    return __builtin_amdgcn_wmma_f32_16x16x4_f32(false, a, false, b,
                                                 (short)0, c, false, false);
}

// ---------------------------------------------------------------------------
// Kernel 1: fused rfft2 + fftshift + mask-gather.  One block per (s,p,c)
// image; all intermediates in LDS (~34KB of 320KB/WGP).  Pass 1: 64-pt row
// DFT to the 33 rfft bins.  Pass 2: 64-pt column DFT evaluated only at the
// K masked (shifted) rows -> feats[sp, :6K] in the reference layout.
// ---------------------------------------------------------------------------
__global__ void fftenc_fft_gather(const float* __restrict__ x,
                                  float* __restrict__ feats) {
    __shared__ float  xs[Hc * 65];        // stride 65: conflict-free column walks
    __shared__ float2 G[Hc * Wr];
    __shared__ float2 tw[64];             // W^k = exp(-i 2*pi*k/64)

    const int tid = threadIdx.x;
    const int bi  = blockIdx.x;           // 0 .. S*P*C-1
    const int sp  = bi / Cd;
    const int cch = bi - sp * Cd;

    if (tid < 64) {
        float a = -2.0f * 3.14159265358979323846f * (float)tid / 64.0f;
        tw[tid] = make_float2(__cosf(a), __sinf(a));
    }

    const float* img = x + (size_t)bi * (Hc * Wc);
    #pragma unroll
    for (int j = 0; j < 4; ++j) {
        int idx = (tid + j * 256) * 4;
        float4 v = *(const float4*)(img + idx);
        int h = idx >> 6, w = idx & 63;
        float* row = xs + h * 65 + w;
        row[0] = v.x; row[1] = v.y; row[2] = v.z; row[3] = v.w;
    }
    __syncthreads();

    for (int t = tid; t < Hc * Wr; t += 256) {       // pass 1 (rows)
        int h = t / Wr, wr = t - h * Wr;
        const float* row = xs + h * 65;
        float gr = 0.f, gi = 0.f;
        int idx = 0;
        #pragma unroll 8
        for (int w = 0; w < 64; ++w) {
            float2 tt = tw[idx];
            idx = (idx + wr) & 63;
            float v = row[w];
            gr += v * tt.x;
            gi += v * tt.y;
        }
        G[t] = make_float2(gr, gi);
    }
    __syncthreads();

    float* frow = feats + (size_t)sp * KPAD16;
    for (int k = tid; k < K; k += 256) {             // pass 2 (masked cols)
        int ix = c_mask.ix[k];
        int hy = (c_mask.iy[k] + 32) & 63;           // fftshift along H
        float fr = 0.f, fi = 0.f;
        int idx = 0;
        #pragma unroll 8
        for (int h = 0; h < 64; ++h) {
            float2 g  = G[h * Wr + ix];
            float2 tt = tw[idx];
            idx = (idx + hy) & 63;
            fr += g.x * tt.x - g.y * tt.y;
            fi += g.x * tt.y + g.y * tt.x;
        }
        frow[cch * K + k]         = fr;
        frow[3 * K + cch * K + k] = fi;
    }

    if (cch == 0 && tid < (KPAD16 - MD))             // zero the K-dim pad
        frow[MD + tid] = 0.f;
}

// ---------------------------------------------------------------------------
// Kernel 2: out(4096,1024) = feats @ proj_w^T + bias, fp32 WMMA.
// Block tile 128M x 64N (8 waves, each a 32x32 tile = 2x2 WMMA tiles).
// Per 16-wide K-step: stage A(128x16)+B(64x16) to LDS with coalesced float4
// loads, then 4 sub-steps x 4 WMMAs from LDS fragments.  LDS row stride
// LDA=20 dwords -> conflict-free ds reads, 16B-aligned stores.  ~21 FLOP/B
// of global (L2-resident) traffic vs 4 FLOP/B for the unblocked version.
// ---------------------------------------------------------------------------
constexpr int LDA = 20;                    // padded LDS row stride (floats)

__global__ void __launch_bounds__(256)
fftenc_wmma_gemm(const float* __restrict__ feats,
                 const float* __restrict__ projw,
                 const float* __restrict__ projb,
                 float* __restrict__ out) {
    __shared__ float Asl[128 * LDA];       // 10.0 KB
    __shared__ float Bsl[ 64 * LDA];       //  5.0 KB

    const int t    = threadIdx.x;
    const int lane = t & 31;
    const int wave = t >> 5;
    const int wm   = wave >> 1;            // 0..3 : 32-row slice
    const int wn   = wave & 1;             // 0..1 : 32-col slice
    const int half = lane >> 4;            // k-pair select (ISA 7.12.2)
    const int l    = lane & 15;

    const int mBase = blockIdx.x * 128;
    const int nBase = blockIdx.y * 64;

    // staging: thread t loads the float4 at (row = t/4, 4*(t%4)) of each slab
    const int sm = t >> 2;                 // 0..63
    const int sq = (t & 3) * 4;            // 0,4,8,12
    const float* aptr0 = feats + (size_t)(mBase + sm)      * KPAD16 + sq;
    const float* aptr1 = feats + (size_t)(mBase + 64 + sm) * KPAD16 + sq;
    const float* bptr  = projw + (size_t)(nBase + sm)      * MD     + sq;
    float* asl0 = &Asl[sm * LDA + sq];
    float* asl1 = &Asl[(64 + sm) * LDA + sq];
    float* bsl  = &Bsl[sm * LDA + sq];

    // per-wave LDS fragment pointers
    const float* af0 = &Asl[(wm * 32      + l) * LDA + half * 2];
    const float* af1 = &Asl[(wm * 32 + 16 + l) * LDA + half * 2];
    const float* bf0 = &Bsl[(wn * 32      + l) * LDA + half * 2];
    const float* bf1 = &Bsl[(wn * 32 + 16 + l) * LDA + half * 2];

    v8f c00 = {}, c01 = {}, c10 = {}, c11 = {};

    for (int kb = 0; kb < KFULL; kb += 16) {
        __syncthreads();                   // previous step's reads done
        float4 va0 = *(const float4*)(aptr0 + kb);
        float4 va1 = *(const float4*)(aptr1 + kb);
        float4 vb  = *(const float4*)(bptr  + kb);
        asl0[0] = va0.x; asl0[1] = va0.y; asl0[2] = va0.z; asl0[3] = va0.w;
        asl1[0] = va1.x; asl1[1] = va1.y; asl1[2] = va1.z; asl1[3] = va1.w;
        bsl[0]  = vb.x;  bsl[1]  = vb.y;  bsl[2]  = vb.z;  bsl[3]  = vb.w;
        __syncthreads();
        #pragma unroll
        for (int ks = 0; ks < 16; ks += 4) {
            v2f a0, a1, b0, b1;
            a0.x = af0[ks]; a0.y = af0[ks + 1];
            a1.x = af1[ks]; a1.y = af1[ks + 1];
            b0.x = bf0[ks]; b0.y = bf0[ks + 1];
            b1.x = bf1[ks]; b1.y = bf1[ks + 1];
            c00 = wmma4(a0, b0, c00);
            c01 = wmma4(a0, b1, c01);
            c10 = wmma4(a1, b0, c10);
            c11 = wmma4(a1, b1, c11);
        }
    }

    if constexpr ((MD & 15) != 0) {        // guarded tail K-step
        __syncthreads();
        float4 va0 = *(const float4*)(aptr0 + KFULL);   // feats zero-padded
        float4 va1 = *(const float4*)(aptr1 + KFULL);
        asl0[0] = va0.x; asl0[1] = va0.y; asl0[2] = va0.z; asl0[3] = va0.w;
        asl1[0] = va1.x; asl1[1] = va1.y; asl1[2] = va1.z; asl1[3] = va1.w;
        #pragma unroll
        for (int j = 0; j < 4; ++j) {
            int d = KFULL + sq + j;
            bsl[j] = (d < MD) ? bptr[KFULL + j] : 0.f;
        }
        __syncthreads();
        #pragma unroll
        for (int ks = 0; ks < 16; ks += 4) {
            v2f a0, a1, b0, b1;
            a0.x = af0[ks]; a0.y = af0[ks + 1];
            a1.x = af1[ks]; a1.y = af1[ks + 1];
            b0.x = bf0[ks]; b0.y = bf0[ks + 1];
            b1.x = bf1[ks]; b1.y = bf1[ks + 1];
            c00 = wmma4(a0, b0, c00);
            c01 = wmma4(a0, b1, c01);
            c10 = wmma4(a1, b0, c10);
            c11 = wmma4(a1, b1, c11);
        }
    }

    // epilogue: C/D layout -> VGPR r: lanes 0-15 M=r, lanes 16-31 M=8+r
    const int n0    = nBase + wn * 32;
    const float bias0 = projb[n0 + l];
    const float bias1 = projb[n0 + 16 + l];
    const int r0 = mBase + wm * 32      + half * 8;
    const int r1 = mBase + wm * 32 + 16 + half * 8;
    #pragma unroll
    for (int r = 0; r < 8; ++r) {
        out[(size_t)(r0 + r) * ENC + n0 + l]      = c00[r] + bias0;
        out[(size_t)(r0 + r) * ENC + n0 + 16 + l] = c01[r] + bias1;
        out[(size_t)(r1 + r) * ENC + n0 + l]      = c10[r] + bias0;
        out[(size_t)(r1 + r) * ENC + n0 + 16 + l] = c11[r] + bias1;
    }
}

// ---------------------------------------------------------------------------
extern "C" void kernel_launch(void* const* d_in, const int* in_sizes, int n_in,
                              void* d_out, int out_size, void* d_ws, size_t ws_size,
                              hipStream_t stream) {
    const float* x  = (const float*)d_in[0];   // (16,256,3,64,64) f32
    // d_in[1] = bc_mask : unused by the reference computation
    const float* pw = (const float*)d_in[2];   // (1024, 6K) f32
    const float* pb = (const float*)d_in[3];   // (1024,) f32
    float* out   = (float*)d_out;              // (4096, 1024) f32
    float* feats = (float*)d_ws;               // (4096, KPAD16) f32 scratch

    fftenc_fft_gather<<<dim3(Sd * Pd * Cd), dim3(256), 0, stream>>>(x, feats);
    fftenc_wmma_gemm<<<dim3((Sd * Pd) / 128, ENC / 64), dim3(256), 0, stream>>>(
        feats, pw, pb, out);
}